// CNNModel_63539746177434
// MI455X (gfx1250) — compile-verified
//
#include <hip/hip_runtime.h>
#include <hip/hip_bf16.h>

// ---------------------------------------------------------------------------
// CDNA5 / gfx1250 forward pass for the sparse-CNN reference.
// All matrix math uses V_WMMA_F32_16X16X4_F32 (exact fp32, wave32 WMMA).
// All WMMA B-operand fetches are branch-free 64-bit vector loads.
// ---------------------------------------------------------------------------

typedef __attribute__((ext_vector_type(2))) float v2f;
typedef __attribute__((ext_vector_type(8))) float v8f;

__device__ __forceinline__ v8f wmma_f32x4(v2f a, v2f b, v8f c) {
  // D = A(16x4,f32) * B(4x16,f32) + C(16x16,f32)
  return __builtin_amdgcn_wmma_f32_16x16x4_f32(
      /*neg_a=*/false, a, /*neg_b=*/false, b,
      /*c_mod=*/(short)0, c, /*reuse_a=*/false, /*reuse_b=*/false);
}

static constexpr int BATCH = 64;
static constexpr int HH = 32;
static constexpr int WW = 32;
static constexpr int DK = HH * WW * 128;   // 131072 flatten length
static constexpr int SPLITS = 128;         // split-K factor for dense1

// ---------------------------------------------------------------------------
// Sparse -> dense im2col weight expansion, emitted directly in the
// K-pair-interleaved layout the WMMA B-fragment wants:
//   wp[((k>>1)*COUT + n)*2 + (k&1)] = Wdense[k][n]
// so one aligned b64 load yields {B[k][n], B[k+1][n]} for a lane.
// Connectivity (circulant Kronecker construction):
//   layer1: channel c feeds node n iff ((n%16)-(c%16)) mod 16 in {0..3}
//   layer2: iff ((n%16)-(c%16)) mod 16 in {0,4,8,12}
// Fan-in slot = block-major (c/16), then rank of (c%16) in the circulant set
// (matches np.nonzero ascending order).
// ---------------------------------------------------------------------------

__global__ void expand_w0(const float* __restrict__ cw0, float* __restrict__ wp) {
  int i = blockIdx.x * blockDim.x + threadIdx.x;
  if (i >= 28 * 64) return;                // K padded 27 -> 28
  int n = i % 64, k = i / 64;
  float v = 0.f;
  if (k < 27) {
    int tap = k / 3, c = k % 3;            // fan-in channels are exactly {0,1,2}
    v = cw0[(tap * 3 + c) * 64 + n];
  }
  wp[(((k >> 1) * 64 + n) << 1) + (k & 1)] = v;
}

__global__ void expand_w1(const float* __restrict__ cw1, float* __restrict__ wp) {
  int i = blockIdx.x * blockDim.x + threadIdx.x;
  if (i >= 576 * 128) return;
  int n = i % 128, k = i / 128;
  int tap = k >> 6, c = k & 63;
  int r = n & 15, cl = c & 15, cb = c >> 4;
  int d = (r - cl) & 15;
  float v = 0.f;
  if (d < 4) {
    int rank = 0;
#pragma unroll
    for (int t = 0; t < 4; ++t) rank += (((r - t) & 15) < cl) ? 1 : 0;
    int f = cb * 4 + rank;                 // fan-in slot (ascending order)
    v = cw1[(tap * 16 + f) * 128 + n];
  }
  wp[(((k >> 1) * 128 + n) << 1) + (k & 1)] = v;
}

__global__ void expand_w2(const float* __restrict__ cw2, float* __restrict__ wp) {
  int i = blockIdx.x * blockDim.x + threadIdx.x;
  if (i >= 1152 * 128) return;
  int n = i % 128, k = i / 128;
  int tap = k >> 7, c = k & 127;
  int r = n & 15, cl = c & 15, cb = c >> 4;
  int d = (r - cl) & 15;
  float v = 0.f;
  if ((d & 3) == 0) {
    int rank = 0;
#pragma unroll
    for (int t = 0; t < 4; ++t) rank += (((r - 4 * t) & 15) < cl) ? 1 : 0;
    int f = cb * 4 + rank;
    v = cw2[(tap * 32 + f) * 128 + n];
  }
  wp[(((k >> 1) * 128 + n) << 1) + (k & 1)] = v;
}

// Pack dw1 [DK,50] into zero-padded pair-interleaved [DK/2][64][2].
__global__ void pack_dw1(const float* __restrict__ dw1, float* __restrict__ wp) {
  int i = blockIdx.x * blockDim.x + threadIdx.x;   // over (DK/2)*64
  if (i >= (DK / 2) * 64) return;
  int n = i & 63;
  int kk = i >> 6;
  float v0 = 0.f, v1 = 0.f;
  if (n < 50) {
    v0 = dw1[(size_t)(2 * kk + 0) * 50 + n];
    v1 = dw1[(size_t)(2 * kk + 1) * 50 + n];
  }
  wp[(size_t)i * 2 + 0] = v0;
  wp[(size_t)i * 2 + 1] = v1;
}

// ---------------------------------------------------------------------------
// Conv layer as im2col GEMM with f32 WMMA.
// Block = 16 consecutive w positions of one (b,h) row; halo staged in LDS.
// Wave w computes the 16x16 output tile for channels [16w, 16w+16).
// ---------------------------------------------------------------------------

template <int CIN>
__device__ __forceinline__ float aelem(const float* __restrict__ tile, int k, int m) {
  if (k >= 9 * CIN) return 0.f;            // only reachable for CIN==3 (K=27)
  int tap = k / CIN;
  int c = k - tap * CIN;
  int dy = tap / 3;
  int dx = tap - dy * 3;
  return tile[(dy * 18 + (m + dx)) * (CIN + 1) + c];
}

template <int CIN, int COUT>
__global__ __launch_bounds__(32 * (COUT / 16))
void conv_wmma(const float* __restrict__ x,      // [B,H,W,CIN]
               const float* __restrict__ wpk,    // pair-interleaved dense weights
               const float* __restrict__ bias,   // [COUT]
               float* __restrict__ y)            // [B,H,W,COUT]
{
  constexpr int K9 = 9 * CIN;
  constexpr int KSTEPS = (K9 + 3) / 4;
  constexpr int LW = CIN + 1;                    // LDS pad vs bank conflicts
  __shared__ float tile[3 * 18 * LW];

  const int blk = blockIdx.x;                    // B*H*(W/16) blocks
  const int w0 = (blk & 1) * 16;
  const int h  = (blk >> 1) & 31;
  const int b  = blk >> 6;

  const int tid = threadIdx.x;
  for (int i = tid; i < 3 * 18 * CIN; i += blockDim.x) {
    int c  = i % CIN;
    int t  = i / CIN;
    int ww = t % 18;
    int dy = t / 18;
    int gh = h + dy - 1;
    int gw = w0 + ww - 1;
    float v = 0.f;
    if (gh >= 0 && gh < HH && gw >= 0 && gw < WW)
      v = x[(((b * HH + gh) * WW) + gw) * CIN + c];
    tile[(dy * 18 + ww) * LW + c] = v;
  }
  __syncthreads();

  const int lane = tid & 31;
  const int wave = tid >> 5;
  const int m    = lane & 15;                    // A: M index; B/C: N index of this lane
  const int half = lane >> 4;                    // K-pair / M+8 selector
  const int nIdx = wave * 16 + m;                // output channel column

  const v2f* __restrict__ wv = (const v2f*)wpk;  // [KSTEPS*2][COUT] pairs

  v8f acc = {};
#pragma unroll 4
  for (int ks = 0; ks < KSTEPS; ++ks) {
    const int ka = ks * 4 + 2 * half;            // K of a.x (a.y = ka+1)
    v2f a;
    a.x = aelem<CIN>(tile, ka, m);
    a.y = aelem<CIN>(tile, ka + 1, m);
    const v2f bb = wv[(ks * 2 + half) * COUT + nIdx];   // {B[ka][n], B[ka+1][n]}
    acc = wmma_f32x4(a, bb, acc);
  }

  const float bv = bias[nIdx];
#pragma unroll
  for (int r = 0; r < 8; ++r) {
    int mo = r + 8 * half;                       // C/D layout: M = r + 8*half
    float v = acc[r] + bv;
    v = v > 0.f ? v : 0.f;                       // relu
    y[(((b * HH + h) * WW) + (w0 + mo)) * COUT + nIdx] = v;
  }
}

// ---------------------------------------------------------------------------
// Dense1: [64, 131072] @ [131072, 50->64] via split-K WMMA + float atomics.
// B comes from the packed pair-interleaved buffer (branch-free b64 loads).
// ---------------------------------------------------------------------------

__global__ __launch_bounds__(32)
void dense1_splitk(const float* __restrict__ x2,   // [64, DK]
                   const float* __restrict__ wpk,  // [DK/2][64][2] packed dw1
                   float* __restrict__ accOut)     // [64, 64] pre-zeroed
{
  const int mt = blockIdx.x;                 // 0..3
  const int nt = blockIdx.y;                 // 0..3
  const int sp = blockIdx.z;                 // 0..SPLITS-1
  const int lane = threadIdx.x;
  const int m = lane & 15;
  const int half = lane >> 4;
  const int row = mt * 16 + m;
  const int n = nt * 16 + m;

  constexpr int KS_PER = DK / 4 / SPLITS;    // 256 wmma steps per split
  const int kbase = sp * (DK / SPLITS);      // even, 8B-aligned fp32 offset
  const float* __restrict__ xr = x2 + (size_t)row * DK + kbase;
  const v2f* __restrict__ wv = (const v2f*)wpk + ((size_t)(kbase >> 1) + half) * 64 + n;

  v8f acc = {};
#pragma unroll 4
  for (int ks = 0; ks < KS_PER; ++ks) {
    const int ka = ks * 4 + 2 * half;
    __builtin_prefetch(xr + ka + 256, 0, 1);          // global_prefetch_b8
    const v2f a = *(const v2f*)(xr + ka);             // {x[row][kg], x[row][kg+1]}
    const v2f b = wv[(size_t)ks * 2 * 64];            // {w[kg][n], w[kg+1][n]}
    acc = wmma_f32x4(a, b, acc);
  }
#pragma unroll
  for (int r = 0; r < 8; ++r) {
    const int mo = mt * 16 + r + 8 * half;
    atomicAdd(&accOut[mo * 64 + nt * 16 + m], acc[r]);
  }
}

__global__ void zero_f32(float* __restrict__ p, int n) {
  int i = blockIdx.x * blockDim.x + threadIdx.x;
  if (i < n) p[i] = 0.f;
}

// ---------------------------------------------------------------------------
// Dense2 (50->10) + softmax. One thread per batch row.
// ---------------------------------------------------------------------------

__global__ __launch_bounds__(64)
void dense2_softmax(const float* __restrict__ acc1,  // [64,64], cols<50 valid
                    const float* __restrict__ db1,
                    const float* __restrict__ dw2,   // [50,10]
                    const float* __restrict__ db2,
                    float* __restrict__ out)         // [64,10]
{
  const int b = threadIdx.x;
  float hbuf[50];
#pragma unroll
  for (int j = 0; j < 50; ++j) {
    float v = acc1[b * 64 + j] + db1[j];
    hbuf[j] = v > 0.f ? v : 0.f;
  }
  float lg[10];
#pragma unroll
  for (int i = 0; i < 10; ++i) {
    float s = db2[i];
    for (int j = 0; j < 50; ++j) s += hbuf[j] * dw2[j * 10 + i];
    lg[i] = s;
  }
  float mx = lg[0];
#pragma unroll
  for (int i = 1; i < 10; ++i) mx = fmaxf(mx, lg[i]);
  float se = 0.f;
#pragma unroll
  for (int i = 0; i < 10; ++i) { lg[i] = expf(lg[i] - mx); se += lg[i]; }
  const float inv = 1.f / se;
#pragma unroll
  for (int i = 0; i < 10; ++i) out[b * 10 + i] = lg[i] * inv;
}

// ---------------------------------------------------------------------------
// Launch
// ---------------------------------------------------------------------------

extern "C" void kernel_launch(void* const* d_in, const int* in_sizes, int n_in,
                              void* d_out, int out_size, void* d_ws, size_t ws_size,
                              hipStream_t stream) {
  const float* inputs = (const float*)d_in[0];   // [64,32,32,3]
  const float* cw0    = (const float*)d_in[1];   // [3,3,3,64]
  const float* cb0    = (const float*)d_in[2];
  const float* cw1    = (const float*)d_in[3];   // [3,3,16,128]
  const float* cb1    = (const float*)d_in[4];
  const float* cw2    = (const float*)d_in[5];   // [3,3,32,128]
  const float* cb2    = (const float*)d_in[6];
  const float* dw1    = (const float*)d_in[7];   // [131072,50]
  const float* db1    = (const float*)d_in[8];
  const float* dw2    = (const float*)d_in[9];   // [50,10]
  const float* db2    = (const float*)d_in[10];
  float* out = (float*)d_out;

  // Workspace carve-up (256B aligned regions).
  char* ws = (char*)d_ws;
  size_t off = 0;
  auto carve = [&](size_t bytes) {
    char* p = ws + off;
    off += (bytes + 255) & ~size_t(255);
    return p;
  };
  float* wd0  = (float*)carve(28 * 64 * sizeof(float));
  float* wd1  = (float*)carve(576 * 128 * sizeof(float));
  float* wd2  = (float*)carve(1152 * 128 * sizeof(float));
  float* acc1 = (float*)carve(64 * 64 * sizeof(float));
  float* bufA = (float*)carve((size_t)BATCH * HH * WW * 128 * sizeof(float));
  float* bufB = (float*)carve((size_t)BATCH * HH * WW * 128 * sizeof(float));
  // bufB doubles as the packed dw1 buffer ((DK/2)*64*2 floats == bufB size)
  // once conv layer 2 has consumed act1.

  // 1) expand sparse conv weights into dense pair-interleaved K x N matrices
  expand_w0<<<(28 * 64 + 255) / 256, 256, 0, stream>>>(cw0, wd0);
  expand_w1<<<(576 * 128 + 255) / 256, 256, 0, stream>>>(cw1, wd1);
  expand_w2<<<(1152 * 128 + 255) / 256, 256, 0, stream>>>(cw2, wd2);
  zero_f32<<<(64 * 64 + 255) / 256, 256, 0, stream>>>(acc1, 64 * 64);

  // 2) conv stack: B*H*(W/16) = 4096 tiles per layer
  const int tiles = BATCH * HH * (WW / 16);
  conv_wmma<3, 64><<<tiles, 32 * (64 / 16), 0, stream>>>(inputs, wd0, cb0, bufA);
  conv_wmma<64, 128><<<tiles, 32 * (128 / 16), 0, stream>>>(bufA, wd1, cb1, bufB);
  conv_wmma<128, 128><<<tiles, 32 * (128 / 16), 0, stream>>>(bufB, wd2, cb2, bufA);

  // 3) pack dw1 into bufB (act1 is dead now), then dense1 split-K WMMA
  pack_dw1<<<((DK / 2) * 64 + 255) / 256, 256, 0, stream>>>(dw1, bufB);
  dense1_splitk<<<dim3(4, 4, SPLITS), 32, 0, stream>>>(bufA, bufB, acc1);

  // 4) dense2 + softmax
  dense2_softmax<<<1, 64, 0, stream>>>(acc1, db1, dw2, db2, out);
  (void)in_sizes; (void)n_in; (void)out_size; (void)ws_size;
}